// FieldConv_6390911336493
// MI455X (gfx1250) — compile-verified
//
#include <hip/hip_runtime.h>

// ---------------- problem constants (match reference) ----------------
#define CI   16
#define CO   32
#define RR   6
#define BB   2
#define MM   5                      // 2*B+1
#define KC   (CI*RR*MM)             // 480 complex per node
#define KR   (2*KC)                 // 960 real
#define NOUT (2*CO)                 // 64 real output cols (Co x {re,im})

typedef float v2f __attribute__((ext_vector_type(2)));
typedef float v8f __attribute__((ext_vector_type(8)));

// ---------------- workspace layout (float offsets) ----------------
// contrib : N * KR                     (76.8 MB, zeroed each call)
// tnode   : N * CI * MM * 2            (12.8 MB)
// WmT     : NOUT * KR                  (245 KB), transposed weight matrix

// =====================================================================
// Kernel 0: expand complex weights U[o,c,r,m] = coeff * exp(i*phase)/M
// into real matrix WmT[64][960] with the (Re,-Im / Im,Re) block layout.
// =====================================================================
__global__ void fc_weights_kernel(const float* __restrict__ zonal,      // (Co,Ci,R)
                                  const float* __restrict__ spherical,  // (Co,Ci,R,B,2)
                                  const float* __restrict__ phase,      // (Co,Ci,B+1)
                                  float* __restrict__ wmt) {            // (64,960)
    int idx = blockIdx.x * blockDim.x + threadIdx.x;
    if (idx >= CO * CI * RR * MM) return;
    int m = idx % MM;
    int r = (idx / MM) % RR;
    int c = (idx / (MM * RR)) % CI;
    int o = idx / (MM * RR * CI);

    // coeff[m] = [conj(sph_b0), conj(sph_b1), zonal, sph_b0, sph_b1]
    float cr, ci;
    if (m == 2) {
        cr = zonal[(o * CI + c) * RR + r];
        ci = 0.0f;
    } else {
        int b = (m < 2) ? m : (m - 3);          // m=0->b0, m=1->b1, m=3->b0, m=4->b1
        const float* s = spherical + (((o * CI + c) * RR + r) * BB + b) * 2;
        cr = s[0];
        ci = (m < 2) ? -s[1] : s[1];            // conjugate for m<2
    }
    // phases[m] = phase[|m-2|]
    int j = (m < 2) ? (2 - m) : (m - 2);
    float p = phase[(o * CI + c) * (BB + 1) + j];
    float sp, cp;
    sincosf(p, &sp, &cp);
    const float inv_m = 1.0f / (float)MM;
    float ur = (cr * cp - ci * sp) * inv_m;     // U = coeff * e^{i p} / M
    float ui = (cr * sp + ci * cp) * inv_m;

    int k = (c * RR + r) * MM + m;              // complex K index, matches contrib layout
    // D = A @ Wm ; WmT[col][kr] = Wm[kr][col]
    wmt[(2 * o)     * KR + 2 * k]     =  ur;
    wmt[(2 * o)     * KR + 2 * k + 1] = -ui;
    wmt[(2 * o + 1) * KR + 2 * k]     =  ui;
    wmt[(2 * o + 1) * KR + 2 * k + 1] =  ur;
}

// =====================================================================
// Kernel 1: T_node[n,c,m] = xc * exp(i*freq_m*phi)  via  u = xc/|xc|
// freqs = [2,1,0,-1,-2]  -> xc*u^2, xc*u, xc, xc*conj(u), xc*conj(u)^2
// =====================================================================
__global__ void fc_tnode_kernel(const float* __restrict__ x,  // (N,Ci,2)
                                float* __restrict__ tnode,    // (N,Ci,M,2)
                                int nc) {
    int i = blockIdx.x * blockDim.x + threadIdx.x;
    if (i >= nc) return;
    float re = x[2 * i], im = x[2 * i + 1];
    float m2 = re * re + im * im;
    float ur = 1.0f, ui = 0.0f;
    if (m2 > 0.0f) {
        float inv = rsqrtf(m2);
        ur = re * inv; ui = im * inv;
    }
    float t1r = re * ur - im * ui,  t1i = re * ui + im * ur;    // xc*u
    float t0r = t1r * ur - t1i * ui, t0i = t1r * ui + t1i * ur; // xc*u^2
    float t3r = re * ur + im * ui,  t3i = im * ur - re * ui;    // xc*conj(u)
    float t4r = t3r * ur + t3i * ui, t4i = t3i * ur - t3r * ui; // xc*conj(u)^2
    float* o = tnode + (long)i * (2 * MM);
    o[0] = t0r; o[1] = t0i;
    o[2] = t1r; o[3] = t1i;
    o[4] = re;  o[5] = im;
    o[6] = t3r; o[7] = t3i;
    o[8] = t4r; o[9] = t4i;
}

// =====================================================================
// Kernel 2: per-edge  T = T_node[src] (x) sten[e]  -> atomic scatter-add
// into contrib[tgt]. One wave per edge. Operands are staged into LDS
// with GLOBAL_LOAD_ASYNC_TO_LDS (CDNA5 async path, ASYNCcnt-tracked) —
// no VGPR round-trip. contrib stays resident in the 192MB L2, so the
// scatter atomics never touch HBM.
// =====================================================================
__global__ void fc_edge_kernel(const float* __restrict__ tnode,   // (N,Ci,M,2)
                               const float* __restrict__ sten,    // (E,R,M,2)
                               const int*   __restrict__ edges,   // (E,2)
                               float* __restrict__ contrib,       // (N,KR)
                               int E) {
    __shared__ float sh[8][224];                  // 160 tnode + 60 sten floats per wave
    int wave = threadIdx.x >> 5;
    int lane = threadIdx.x & 31;
    int e = blockIdx.x * 8 + wave;

    int src = 0, tgt = 0;
    if (e < E) {
        src = edges[2 * e];
        tgt = edges[2 * e + 1];
        const float* gt = tnode + (long)src * (CI * MM * 2);
        const float* gs = sten + (long)e * (RR * MM * 2);
        // memory -> LDS, bypassing VGPRs; each b32 op moves 128B per wave
        for (int i = lane; i < CI * MM * 2; i += 32) {      // 5 ops
            unsigned lds = (unsigned)(size_t)&sh[wave][i];  // low 32b of flat LDS addr
            asm volatile("global_load_async_to_lds_b32 %0, %1, off"
                         :: "v"(lds), "v"((unsigned long long)(size_t)(gt + i))
                         : "memory");
        }
        for (int i = lane; i < RR * MM * 2; i += 32) {      // 2 ops (2nd partially masked)
            unsigned lds = (unsigned)(size_t)&sh[wave][160 + i];
            asm volatile("global_load_async_to_lds_b32 %0, %1, off"
                         :: "v"(lds), "v"((unsigned long long)(size_t)(gs + i))
                         : "memory");
        }
    }
    // drain our wave's async copies, then block-barrier so every wave's
    // LDS region is visible (compiler barrier does not cover ASYNCcnt)
    asm volatile("s_wait_asynccnt 0x0" ::: "memory");
    __syncthreads();

    if (e < E) {
        const float* tn = sh[wave];
        const float* st = sh[wave] + 160;
        float* dst = contrib + (long)tgt * KR;
        #pragma unroll
        for (int k = lane; k < KC; k += 32) {     // 15 complex MACs per lane
            int c  = k / (RR * MM);
            int rm = k - c * (RR * MM);
            int r  = rm / MM;
            int m  = rm - r * MM;
            float ar = tn[(c * MM + m) * 2], ai = tn[(c * MM + m) * 2 + 1];
            float br = st[(r * MM + m) * 2], bi = st[(r * MM + m) * 2 + 1];
            atomicAdd(dst + 2 * k,     ar * br - ai * bi);
            atomicAdd(dst + 2 * k + 1, ar * bi + ai * br);
        }
    }
}

// =====================================================================
// Kernel 3: D[20000 x 64] = contrib[20000 x 960] @ Wm[960 x 64]
// fp32 WMMA 16x16x4. 8 waves/block: 2 node-subtiles x 4 col-tiles.
// D rows are exactly the flat (Co,2) output rows -> write d_out directly.
// =====================================================================
__global__ void fc_gemm_kernel(const float* __restrict__ A,   // contrib (N,KR)
                               const float* __restrict__ Bt,  // WmT (64,KR)
                               float* __restrict__ D,         // out (N,64)
                               int Nn) {
    int wave    = threadIdx.x >> 5;
    int lane    = threadIdx.x & 31;
    int colTile = wave & 3;                 // 4 tiles of 16 output cols
    int nodeSub = wave >> 2;                // 2 tiles of 16 nodes
    int nodeBase = blockIdx.x * 32 + nodeSub * 16;
    if (nodeBase >= Nn) return;             // wave-uniform; EXEC stays all-ones

    int hi = lane >> 4;                     // 0: K pair {0,1}, 1: K pair {2,3}
    int lo = lane & 15;
    // A frag: lane holds A[node=lo][k + 2*hi .. +1]   (ISA 16x4 f32 layout)
    const float* arow = A + (long)(nodeBase + lo) * KR + 2 * hi;
    // B frag: lane holds B[k + 2*hi][col], B[k+1+2*hi][col]  (Bt row-contig)
    int col = colTile * 16 + lo;
    const float* brow = Bt + (long)col * KR + 2 * hi;

    v8f acc = {};
    for (int k = 0; k < KR; k += 4) {
        if ((k & 63) == 0)
            __builtin_prefetch(arow + k + 256, 0, 3);   // WGP-scope global_prefetch_b8
        v2f a = *(const v2f*)(arow + k);
        v2f b = *(const v2f*)(brow + k);
        acc = __builtin_amdgcn_wmma_f32_16x16x4_f32(
                  false, a, false, b, (short)0, acc, false, false);
    }
    // C/D layout: VGPR v holds row M = v + 8*hi, col = lane&15
    float* drow = D + (long)nodeBase * NOUT + colTile * 16 + lo;
    #pragma unroll
    for (int v = 0; v < 8; ++v) {
        drow[(long)(v + 8 * hi) * NOUT] = acc[v];
    }
}

// =====================================================================
extern "C" void kernel_launch(void* const* d_in, const int* in_sizes, int n_in,
                              void* d_out, int out_size, void* d_ws, size_t ws_size,
                              hipStream_t stream) {
    const float* x      = (const float*)d_in[0];   // (N,Ci,2)
    const float* sten   = (const float*)d_in[1];   // (E,R,M,2)
    const float* zonal  = (const float*)d_in[2];   // (Co,Ci,R)
    const float* sph    = (const float*)d_in[3];   // (Co,Ci,R,B,2)
    const float* phase  = (const float*)d_in[4];   // (Co,Ci,B+1)
    const int*   edges  = (const int*)d_in[5];     // (E,2)
    float* out = (float*)d_out;

    const int N = in_sizes[0] / (CI * 2);
    const int E = in_sizes[5] / 2;

    float* ws      = (float*)d_ws;
    float* contrib = ws;                                  // N*KR
    float* tnode   = ws + (size_t)N * KR;                 // N*Ci*M*2
    float* wmt     = tnode + (size_t)N * CI * MM * 2;     // 64*KR

    // weights (independent)
    {
        int total = CO * CI * RR * MM;
        fc_weights_kernel<<<(total + 255) / 256, 256, 0, stream>>>(zonal, sph, phase, wmt);
    }
    // per-node rotations (independent)
    {
        int nc = N * CI;
        fc_tnode_kernel<<<(nc + 255) / 256, 256, 0, stream>>>(x, tnode, nc);
    }
    // zero the scatter target (must happen every call; stream-ordered)
    hipMemsetAsync(contrib, 0, (size_t)N * KR * sizeof(float), stream);
    // edge scatter (one wave per edge, async-to-LDS staging)
    {
        int blocks = (E + 7) / 8;
        fc_edge_kernel<<<blocks, 256, 0, stream>>>(tnode, sten, edges, contrib, E);
    }
    // fp32 WMMA node contraction, writes d_out directly
    {
        int blocks = (N + 31) / 32;
        fc_gemm_kernel<<<blocks, 256, 0, stream>>>(contrib, wmt, out, N);
    }
}